// GaussianVecIou_35416300323281
// MI455X (gfx1250) — compile-verified
//
#include <hip/hip_runtime.h>
#include <math.h>

typedef float v4f __attribute__((ext_vector_type(4)));
typedef int   v4i __attribute__((ext_vector_type(4)));

#define TI 32
#define TJ 128

// ---- CDNA5 async global->LDS copy (16B per lane), with portable fallback ----
__device__ __forceinline__ void async_b128(const float* g, float* l) {
#if __has_builtin(__builtin_amdgcn_global_load_async_to_lds_b128)
  __builtin_amdgcn_global_load_async_to_lds_b128(
      (__attribute__((address_space(1))) v4i*)g,
      (__attribute__((address_space(3))) v4i*)l,
      /*imm offset*/0, /*cpol*/0);
#else
  *(v4f*)l = *(const v4f*)g;
#endif
}

__device__ __forceinline__ void wait_async_then_barrier() {
#if __has_builtin(__builtin_amdgcn_global_load_async_to_lds_b128)
#if __has_builtin(__builtin_amdgcn_s_wait_asynccnt)
  __builtin_amdgcn_s_wait_asynccnt(0);
#else
  asm volatile("s_wait_asynccnt 0" ::: "memory");
#endif
#endif
  __syncthreads();
}

// ---- Pass 1: per-item record {cx, cy, s00, s01, s11, q', pad, pad} ----
// q' = 0.25*log(det_i) + 0.5*ln2  (absorbs the -ln2 from using det4 = 4*det in pass 2)
__global__ __launch_bounds__(256)
void precomp_kernel(const float* __restrict__ vecs,
                    const float* __restrict__ thick,
                    const float* __restrict__ mask,
                    float* __restrict__ rec, int n_items)
{
  int idx = blockIdx.x * 256 + threadIdx.x;
  if (idx >= n_items) return;
  v4f v = ((const v4f*)vecs)[idx];           // x0,y0,x1,y1
  float th = thick[idx];
  float mk = mask[idx];
  float cx = 0.5f * (v.x + v.z);
  float cy = 0.5f * (v.y + v.w);
  float dx = v.x - v.z;
  float dy = v.y - v.w;
  float len = sqrtf(dx * dx + dy * dy);
  const float IS2 = 0.70710678118654752440f; // 1/sqrt(2)
  float om = 1.0f - mk;
  float sx = th * mk + len * IS2 * om;       // bbox_size[0]
  float sy = len * mk + len * IS2 * om;      // bbox_size[1]
  float ang = atan2f(dx, dy) * mk;           // arctan2(diff_x, diff_y)*mask
  float c = cosf(ang), s = sinf(ang);
  float a = 0.5f * sx, b = 0.5f * sy;        // diag(delta)
  float c2 = c * c, s2 = s * s, cs = c * s;
  float s00 = a * c2 + b * s2;               // R*diag*R^T
  float s01 = (a - b) * cs;
  float s11 = a * s2 + b * c2;
  float det = s00 * s11 - s01 * s01;
  const float HALF_LN2 = 0.34657359027997265f;
  float q = fmaf(0.25f, __logf(det), HALF_LN2);
  v4f r0 = {cx, cy, s00, s01};
  v4f r1 = {s11, q, 0.0f, 0.0f};
  v4f* r = (v4f*)rec;
  r[(size_t)idx * 2 + 0] = r0;
  r[(size_t)idx * 2 + 1] = r1;
}

// ---- Pass 2: pairwise Gaussian IoU over 32x128 tiles, j/i records staged in LDS ----
__global__ __launch_bounds__(256)
void pair_kernel(const float* __restrict__ rec, float* __restrict__ out,
                 int N, int tiN, int tjN)
{
  __shared__ float sj[TJ * 8];   // 4 KB
  __shared__ float si[TI * 8];   // 1 KB

  int blk = blockIdx.x;
  int tj = blk % tjN;
  int tmp = blk / tjN;
  int ti = tmp % tiN;
  int b  = tmp / tiN;
  int i0 = ti * TI, j0 = tj * TJ;
  int t = threadIdx.x;

  const float* gj = rec + ((size_t)b * N + j0) * 8;
  const float* gi = rec + ((size_t)b * N + i0) * 8;
  int nvj = (N - j0 < TJ) ? (N - j0) : TJ;
  int nvi = (N - i0 < TI) ? (N - i0) : TI;

  if (t < nvj * 2) async_b128(gj + t * 4, sj + t * 4);   // 256 x 16B = 4KB
  if (t < nvi * 2) async_b128(gi + t * 4, si + t * 4);   // 64 x 16B  = 1KB
  wait_async_then_barrier();

  const v4f* sj4 = (const v4f*)sj;
  const v4f* si4 = (const v4f*)si;

  int jj  = (t & 31) * 4;   // 4 consecutive j per thread; a wave covers 128 cols
  int iof = t >> 5;         // wave id = row offset within tile
  bool jfull = (jj + 3) < nvj;

#pragma unroll
  for (int r = 0; r < TI / 8; ++r) {
    int il = iof + r * 8;
    if (il >= nvi) break;
    int i = i0 + il;
    v4f ia = si4[il * 2], ib = si4[il * 2 + 1];
    float cxi = ia.x, cyi = ia.y, s00i = ia.z, s01i = ia.w;
    float s11i = ib.x, qi = ib.y;
    v4f res;
#pragma unroll
    for (int k = 0; k < 4; ++k) {
      int jl = jj + k;
      v4f ja = sj4[jl * 2], jb = sj4[jl * 2 + 1];
      float ddx = cxi - ja.x;
      float ddy = cyi - ja.y;
      // Unhalved sums: t = sigma_i + sigma_j = 2*m;  det4 = 4*det(m)
      float t00 = s00i + ja.z;
      float t01 = s01i + ja.w;
      float t11 = s11i + jb.x;
      float det4 = fmaf(t00, t11, -(t01 * t01));
      float invd = (det4 != 0.0f) ? __builtin_amdgcn_rcpf(det4) : 0.0f;
      // quad = (ddx^2*m11 + ddy^2*m00)/(8*det(m)) = (ddx^2*t11 + ddy^2*t00)*0.25/det4
      float num = fmaf(ddx * ddx, t11, ddy * ddy * t00);
      // Bd = quad + 0.5*log(det) - (0.25*log(det_i)+0.25*log(det_j))
      //    = num*0.25*invd + 0.5*log(det4) - (qi + qj)     [ln2 folded into q']
      float Bd = fmaf(num, 0.25f * invd, fmaf(0.5f, __logf(det4), -(qi + jb.y)));
      float Bc = __expf(-Bd);
      float Hd = __builtin_amdgcn_sqrtf(1.0f - Bc);
      res[k] = 1.0f - Hd;
    }
    float* orow = out + (((size_t)b * N + i) * N + j0 + jj);
    if (jfull) {
      *(v4f*)orow = res;                     // 16B store; wave writes 512B contiguous
    } else {
      for (int k = 0; k < 4; ++k)
        if (jj + k < nvj) orow[k] = res[k];
    }
  }
}

extern "C" void kernel_launch(void* const* d_in, const int* in_sizes, int n_in,
                              void* d_out, int out_size, void* d_ws, size_t ws_size,
                              hipStream_t stream) {
  const float* vecs  = (const float*)d_in[0];
  const float* thick = (const float*)d_in[1];
  const float* mask  = (const float*)d_in[2];
  float* out = (float*)d_out;
  float* rec = (float*)d_ws;                  // n_items * 8 floats (256 KB)

  int n_items = in_sizes[1];                  // B*N (thickness is (B,N,1))
  int N = (int)((long long)out_size / n_items);
  int B = n_items / N;

  int blocks1 = (n_items + 255) / 256;
  hipLaunchKernelGGL(precomp_kernel, dim3(blocks1), dim3(256), 0, stream,
                     vecs, thick, mask, rec, n_items);

  int tiN = (N + TI - 1) / TI;
  int tjN = (N + TJ - 1) / TJ;
  hipLaunchKernelGGL(pair_kernel, dim3(B * tiN * tjN), dim3(256), 0, stream,
                     rec, out, N, tiN, tjN);
}